// RFM_24464133718900
// MI455X (gfx1250) — compile-verified
//
#include <hip/hip_runtime.h>
#include <hip/hip_bf16.h>

// NAT/Restormer block on MI455X (gfx1250). All tensors fp32, NCHW (C x P,
// P = 192*192). 1x1 convs / projections = GEMMs via V_WMMA_F32_16X16X4_F32.
// Weight tile staged in LDS via GLOBAL_LOAD_ASYNC_TO_LDS_B32 (+ s_wait_asynccnt),
// A-fragments read with ds_load_b64. All fragment loads are branch-free
// (clamped addresses + v_cndmask zeroing) so EXEC stays all-ones into WMMA.
// Workspace (floats): A[340*P] B[170*P] C/D/E[64*P] ~= 104 MB (fits 192MB L2).

#define DEVFN __device__ __forceinline__

constexpr int HH = 192, WW = 192, P = HH * WW;  // 36864 positions

typedef __attribute__((ext_vector_type(2))) float v2f;
typedef __attribute__((ext_vector_type(8))) float v8f;

// ---------------- depthwise 3x3, SAME padding ----------------
DEVFN float dw3x3(const float* __restrict__ s, const float* __restrict__ w,
                  int i, int j) {
  float acc = 0.f;
#pragma unroll
  for (int dy = -1; dy <= 1; ++dy) {
    int y = i + dy;
    if ((unsigned)y >= (unsigned)HH) continue;
#pragma unroll
    for (int dx = -1; dx <= 1; ++dx) {
      int x = j + dx;
      if ((unsigned)x >= (unsigned)WW) continue;
      acc = fmaf(w[(dy + 1) * 3 + (dx + 1)], s[y * WW + x], acc);
    }
  }
  return acc;
}

DEVFN float gelu_exact(float x) {
  return 0.5f * x * (1.f + erff(x * 0.70710678118654752f));
}

// ---------------- embed: concat + depthwise 3x3 (192ch) + bias --------------
__global__ void embed_dw_kernel(const float* __restrict__ x1,
                                const float* __restrict__ x2,
                                const float* __restrict__ x3,
                                const float* __restrict__ wdw,
                                const float* __restrict__ bdw,
                                float* __restrict__ out) {
  int idx = blockIdx.x * blockDim.x + threadIdx.x;
  if (idx >= 192 * P) return;
  int c = idx / P, p = idx - c * P;
  const float* src =
      (c < 64 ? x1 : (c < 128 ? x2 : x3)) + (size_t)(c & 63) * P;
  out[idx] = dw3x3(src, wdw + c * 9, p / WW, p % WW) + bdw[c];
}

// ---------------- bias-free LayerNorm over 64 channels ----------------
__global__ void ln_kernel(const float* __restrict__ x,
                          const float* __restrict__ w,
                          float* __restrict__ y) {
  int p = blockIdx.x * blockDim.x + threadIdx.x;
  if (p >= P) return;
  float s = 0.f, s2 = 0.f;
#pragma unroll 8
  for (int c = 0; c < 64; ++c) {
    float v = x[(size_t)c * P + p];
    s += v;
    s2 = fmaf(v, v, s2);
  }
  float mean = s * (1.f / 64.f);
  float var = s2 * (1.f / 64.f) - mean * mean;
  float inv = rsqrtf(var + 1e-5f);
#pragma unroll 8
  for (int c = 0; c < 64; ++c)
    y[(size_t)c * P + p] = x[(size_t)c * P + p] * inv * w[c];
}

// ---------------- GEMM: O[MxP] = Wm[MxKD] * X[KDxP] (+bias) ----------------
// Block = 4 waves; each wave computes two 16x16 output tiles (shared A-frag).
// Weight tile (16 x KD) staged into LDS with async global->LDS copies.
// A frag (16x4): lane%16 = M row; K = (lane/16)*2 + vgpr   (ds_load_b64)
// B frag (4x16): lane%16 = N col; K = (lane/16)*2 + vgpr
// C/D (16x16):   VGPR r -> M = r + 8*(lane/16), N = lane%16
template <int KD>
__global__ void __launch_bounds__(128)
wmma_gemm_nchw(const float* __restrict__ Wm, const float* __restrict__ X,
               const float* __restrict__ bias, float* __restrict__ O, int M) {
  __shared__ float Ws[16 * KD + 8];
  const int lane = threadIdx.x & 31;
  const int wave = threadIdx.x >> 5;
  const int mtile = blockIdx.y * 16;

  // ---- async-stage 16 x KD weight tile into LDS (ASYNCcnt path) ----
  {
    const unsigned lds_base = (unsigned)(uintptr_t)Ws;  // low 32b = LDS offset
    for (int t = threadIdx.x; t < 16 * KD; t += 128) {
      int r = t / KD, k = t - r * KD;
      int mm = mtile + r;
      if (mm >= M) mm = M - 1;  // clamp: duplicated rows are masked at store
      const float* gsrc = Wm + (size_t)mm * KD + k;
      unsigned ldst = lds_base + (unsigned)t * 4u;
      asm volatile("global_load_async_to_lds_b32 %0, %1, off"
                   :
                   : "v"(ldst), "v"(gsrc)
                   : "memory");
    }
    asm volatile("s_wait_asynccnt 0x0" ::: "memory");
  }
  __syncthreads();

  const int pbase = (blockIdx.x * 4 + wave) * 32 + (lane & 15);
  const int khalf = (lane >> 4) << 1;  // 0 or 2
  const float* Xc = X + pbase;
  const float* wrow = &Ws[(lane & 15) * KD];

  v8f acc0 = {}, acc1 = {};
#pragma unroll 8
  for (int k0 = 0; k0 < KD; k0 += 4) {
    const int ka = k0 + khalf;  // even
    const int kc0 = (ka < KD) ? ka : (KD - 1);
    const int kc1 = (ka + 1 < KD) ? (ka + 1) : (KD - 1);
    float2 af = *(const float2*)&wrow[ka];  // ds_load_b64 (in-bounds, padded)
    v2f a, b0, b1;
    a.x = (ka < KD) ? af.x : 0.f;
    a.y = (ka + 1 < KD) ? af.y : 0.f;
    float x00 = Xc[(size_t)kc0 * P];
    float x01 = Xc[(size_t)kc1 * P];
    float x10 = Xc[(size_t)kc0 * P + 16];
    float x11 = Xc[(size_t)kc1 * P + 16];
    b0.x = (ka < KD) ? x00 : 0.f;
    b0.y = (ka + 1 < KD) ? x01 : 0.f;
    b1.x = (ka < KD) ? x10 : 0.f;
    b1.y = (ka + 1 < KD) ? x11 : 0.f;
    acc0 = __builtin_amdgcn_wmma_f32_16x16x4_f32(false, a, false, b0, (short)0,
                                                 acc0, false, false);
    acc1 = __builtin_amdgcn_wmma_f32_16x16x4_f32(false, a, false, b1, (short)0,
                                                 acc1, false, false);
  }
#pragma unroll
  for (int r = 0; r < 8; ++r) {
    int mm = mtile + r + ((lane >> 4) << 3);
    if (mm < M) {
      float bv = bias ? bias[mm] : 0.f;
      O[(size_t)mm * P + pbase] = acc0[r] + bv;
      O[(size_t)mm * P + pbase + 16] = acc1[r] + bv;
    }
  }
}

// ---------------- dilated neighborhood attention (K=7, dil=3, hd=32) -------
DEVFN void win_start(int idx, int& s, int& pb) {
  // length=192, K=7, dil=3, nh=3; 192 % 3 == 0 -> b == 0 branch of reference
  int ni = idx - 9;
  int imodd = idx % 3;
  if (ni < 0) {
    s = imodd;
    pb = 6 - idx / 3;
  } else if (idx + 9 >= HH) {
    s = 171 + imodd;  // a + imodd - K*dil = 192 + imodd - 21
    pb = (HH - 1 - idx) / 3;
  } else {
    s = ni;
    pb = 3;
  }
}

__global__ void na2d_kernel(const float* __restrict__ qkv,  // 192 x P (NCHW)
                            const float* __restrict__ rpb,  // 2 x 13 x 13
                            float* __restrict__ out) {      // 64 x P
  int idx = blockIdx.x * blockDim.x + threadIdx.x;
  if (idx >= 2 * P) return;
  int head = idx & 1;
  int p = idx >> 1;
  int i = p / WW, j = p - (p / WW) * WW;
  int sh, pbh, sw, pbw;
  win_start(i, sh, pbh);
  win_start(j, sw, pbw);

  const float* qb = qkv + (size_t)(head * 32) * P + p;
  const float* kb = qkv + (size_t)(64 + head * 32) * P;
  const float* vb = qkv + (size_t)(128 + head * 32) * P;
  const float* rb = rpb + head * 169;

  float q[32];
#pragma unroll
  for (int d = 0; d < 32; ++d) q[d] = qb[(size_t)d * P] * 0.17677669529663687f;

  float m = -1e30f, ssum = 0.f;
  float o[32];
#pragma unroll
  for (int d = 0; d < 32; ++d) o[d] = 0.f;

  for (int ky = 0; ky < 7; ++ky) {
    int ih = sh + ky * 3;
    for (int kx = 0; kx < 7; ++kx) {
      int iw = sw + kx * 3;
      int pn = ih * WW + iw;
      float dot = 0.f;
#pragma unroll
      for (int d = 0; d < 32; ++d) dot = fmaf(q[d], kb[(size_t)d * P + pn], dot);
      dot += rb[(pbh + ky) * 13 + (pbw + kx)];
      float mn = fmaxf(m, dot);
      float sc = __expf(m - mn);
      float e = __expf(dot - mn);
      ssum = ssum * sc + e;
#pragma unroll
      for (int d = 0; d < 32; ++d)
        o[d] = fmaf(e, vb[(size_t)d * P + pn], o[d] * sc);
      m = mn;
    }
  }
  float inv = 1.f / ssum;
  float* ob = out + (size_t)(head * 32) * P + p;
#pragma unroll
  for (int d = 0; d < 32; ++d) ob[(size_t)d * P] = o[d] * inv;
}

// ---------------- fusion gate: dw3x3 on concat(en,de), gelu(x1)*x2 ----------
__global__ void fusion_gate_kernel(const float* __restrict__ en,
                                   const float* __restrict__ de,
                                   const float* __restrict__ wdw,  // 128x9
                                   float* __restrict__ g) {        // 64 x P
  int idx = blockIdx.x * blockDim.x + threadIdx.x;
  if (idx >= 64 * P) return;
  int c = idx / P, p = idx - c * P;
  int i = p / WW, j = p % WW;
  float t1 = dw3x3(en + (size_t)c * P, wdw + c * 9, i, j);
  float t2 = dw3x3(de + (size_t)c * P, wdw + (64 + c) * 9, i, j);
  g[idx] = gelu_exact(t1) * t2;
}

// ---------------- ffn gate: dw3x3 on 340ch, split 170/170, gelu gate --------
__global__ void ffn_gate_kernel(const float* __restrict__ t,    // 340 x P
                                const float* __restrict__ wdw,  // 340x9
                                float* __restrict__ g) {        // 170 x P
  int idx = blockIdx.x * blockDim.x + threadIdx.x;
  if (idx >= 170 * P) return;
  int c = idx / P, p = idx - c * P;
  int i = p / WW, j = p % WW;
  float t1 = dw3x3(t + (size_t)c * P, wdw + c * 9, i, j);
  float t2 = dw3x3(t + (size_t)(170 + c) * P, wdw + (170 + c) * 9, i, j);
  g[idx] = gelu_exact(t1) * t2;
}

extern "C" void kernel_launch(void* const* d_in, const int* in_sizes, int n_in,
                              void* d_out, int out_size, void* d_ws,
                              size_t ws_size, hipStream_t stream) {
  const float* x1 = (const float*)d_in[0];
  const float* x2 = (const float*)d_in[1];
  const float* x3 = (const float*)d_in[2];
  const float* emb_dw_w = (const float*)d_in[3];
  const float* emb_dw_b = (const float*)d_in[4];
  const float* emb_pw_w = (const float*)d_in[5];
  const float* emb_pw_b = (const float*)d_in[6];
  const float* norm1_w = (const float*)d_in[7];
  const float* norm2_w = (const float*)d_in[8];
  const float* qkv_w = (const float*)d_in[9];
  const float* qkv_b = (const float*)d_in[10];
  const float* rpb = (const float*)d_in[11];
  const float* proj_w = (const float*)d_in[12];
  const float* proj_b = (const float*)d_in[13];
  const float* fus_dw_w = (const float*)d_in[14];
  const float* fus_out_w = (const float*)d_in[15];
  const float* ffn_in_w = (const float*)d_in[16];
  const float* ffn_dw_w = (const float*)d_in[17];
  const float* ffn_out_w = (const float*)d_in[18];

  float* ws = (float*)d_ws;
  float* A = ws;                    // 340*P  (t192 / qkv / t340)
  float* B = A + (size_t)340 * P;   // 170*P  (x0 / g2)
  float* C = B + (size_t)170 * P;   // 64*P   (yln / attn / gate / z)
  float* D = C + (size_t)64 * P;    // 64*P   (na out-proj / y2)
  float* E = D + (size_t)64 * P;    // 64*P   (x after fusion1)

  const int TB = 256;
  auto nblk = [](long long n) { return dim3((unsigned)((n + 255) / 256)); };
  const dim3 gblk(128);
  const unsigned GX = P / 128;  // 4 waves x 2 tiles x 16 cols per block

  // 1) embed depthwise + bias -> A (192 x P)
  embed_dw_kernel<<<nblk((long long)192 * P), TB, 0, stream>>>(
      x1, x2, x3, emb_dw_w, emb_dw_b, A);
  // 2) pointwise 192->64 + bias -> B (x0)
  wmma_gemm_nchw<192><<<dim3(GX, 4), gblk, 0, stream>>>(emb_pw_w, A, emb_pw_b,
                                                        B, 64);
  // 3) LN1 -> C
  ln_kernel<<<nblk(P), TB, 0, stream>>>(B, norm1_w, C);
  // 4) qkv 64->192 + bias -> A
  wmma_gemm_nchw<64><<<dim3(GX, 12), gblk, 0, stream>>>(qkv_w, C, qkv_b, A,
                                                        192);
  // 5) dilated neighborhood attention -> C (64 x P)
  na2d_kernel<<<nblk((long long)2 * P), TB, 0, stream>>>(A, rpb, C);
  // 6) out projection 64->64 + bias -> D
  wmma_gemm_nchw<64><<<dim3(GX, 4), gblk, 0, stream>>>(proj_w, C, proj_b, D,
                                                       64);
  // 7) fusion #1: gate(en=B, de=D) -> C ; 1x1 64->64 -> E
  fusion_gate_kernel<<<nblk((long long)64 * P), TB, 0, stream>>>(B, D,
                                                                 fus_dw_w, C);
  wmma_gemm_nchw<64><<<dim3(GX, 4), gblk, 0, stream>>>(fus_out_w, C, nullptr,
                                                       E, 64);
  // 8) LN2 -> C
  ln_kernel<<<nblk(P), TB, 0, stream>>>(E, norm2_w, C);
  // 9) ffn in 64->340 -> A
  wmma_gemm_nchw<64><<<dim3(GX, 22), gblk, 0, stream>>>(ffn_in_w, C, nullptr,
                                                        A, 340);
  // 10) ffn depthwise + gate -> B (170 x P)
  ffn_gate_kernel<<<nblk((long long)170 * P), TB, 0, stream>>>(A, ffn_dw_w, B);
  // 11) ffn out 170->64 -> D
  wmma_gemm_nchw<170><<<dim3(GX, 4), gblk, 0, stream>>>(ffn_out_w, B, nullptr,
                                                        D, 64);
  // 12) fusion #2: gate(en=E, de=D) -> C ; 1x1 64->64 -> d_out
  fusion_gate_kernel<<<nblk((long long)64 * P), TB, 0, stream>>>(E, D,
                                                                 fus_dw_w, C);
  wmma_gemm_nchw<64><<<dim3(GX, 4), gblk, 0, stream>>>(fus_out_w, C, nullptr,
                                                       (float*)d_out, 64);
}